// localHead_74783970558281
// MI455X (gfx1250) — compile-verified
//
#include <hip/hip_runtime.h>
#include <hip/hip_bf16.h>

typedef __attribute__((ext_vector_type(2))) float v2f;
typedef __attribute__((ext_vector_type(4))) float v4f;
typedef __attribute__((ext_vector_type(8))) float v8f;

#define Bb 8
#define Cc 256
#define Hh 128
#define Ww 128
#define Nn 4096
#define Ee 128
#define DESC_PITCH 260   // 260 % 64 == 4 -> conflict-free b64 A-fragment loads

__global__ __launch_bounds__(256)
void localhead_fused_kernel(const float* __restrict__ x,
                            const float* __restrict__ kpts,
                            const float* __restrict__ ww,
                            const float* __restrict__ wb,
                            float* __restrict__ out)
{
    __shared__ float sDesc[16 * DESC_PITCH];  // 16.25 KB sampled descriptors
    __shared__ float sOut[16 * Ee];           // 8 KB whitened tile
    __shared__ float sWgt[16][4];             // bilinear weights (masked)
    __shared__ int   sOff[16][4];             // clamped texel offsets
    __shared__ float sPart[16 * 16];
    __shared__ float sRn[16];

    const int t    = threadIdx.x;
    const int tile = blockIdx.x;              // 2048 tiles = 8 batches * 256 row-tiles
    const int b    = tile >> 8;
    const int n0   = (tile & 255) << 4;       // first of 16 points in this tile

    // ---- Phase 0: per-point bilinear setup (grid_sample, zeros pad, align_corners=False)
    if (t < 16) {
        const size_t kbase = ((size_t)(b * Nn + n0 + t)) * 2;
        const float kx = kpts[kbase + 0];
        const float ky = kpts[kbase + 1];
        const float ix = (kx + 1.0f) * (Ww * 0.5f) - 0.5f;
        const float iy = (ky + 1.0f) * (Hh * 0.5f) - 0.5f;
        const float x0f = floorf(ix), y0f = floorf(iy);
        const float wx = ix - x0f, wy = iy - y0f;
        const int x0 = (int)x0f, y0 = (int)y0f;
        const int x1 = x0 + 1,   y1 = y0 + 1;
        const float mx0 = (x0 >= 0 && x0 < Ww) ? 1.0f : 0.0f;
        const float mx1 = (x1 >= 0 && x1 < Ww) ? 1.0f : 0.0f;
        const float my0 = (y0 >= 0 && y0 < Hh) ? 1.0f : 0.0f;
        const float my1 = (y1 >= 0 && y1 < Hh) ? 1.0f : 0.0f;
        const int xc0 = min(max(x0, 0), Ww - 1), xc1 = min(max(x1, 0), Ww - 1);
        const int yc0 = min(max(y0, 0), Hh - 1), yc1 = min(max(y1, 0), Hh - 1);
        sWgt[t][0] = (1.0f - wx) * (1.0f - wy) * mx0 * my0;
        sWgt[t][1] = wx * (1.0f - wy) * mx1 * my0;
        sWgt[t][2] = (1.0f - wx) * wy * mx0 * my1;
        sWgt[t][3] = wx * wy * mx1 * my1;
        sOff[t][0] = yc0 * Ww + xc0;
        sOff[t][1] = yc0 * Ww + xc1;
        sOff[t][2] = yc1 * Ww + xc0;
        sOff[t][3] = yc1 * Ww + xc1;
    }
    __syncthreads();

    // ---- Phase 1: gather+blend, one channel per thread (L2-resident feature maps)
    {
        const int c = t;
        const float* __restrict__ plane = x + ((size_t)b * Cc + c) * (size_t)(Hh * Ww);
        #pragma unroll 4
        for (int pt = 0; pt < 16; ++pt) {
            const float w00 = sWgt[pt][0], w10 = sWgt[pt][1];
            const float w01 = sWgt[pt][2], w11 = sWgt[pt][3];
            const int o00 = sOff[pt][0], o10 = sOff[pt][1];
            const int o01 = sOff[pt][2], o11 = sOff[pt][3];
            float v = w00 * plane[o00];
            v = fmaf(w10, plane[o10], v);
            v = fmaf(w01, plane[o01], v);
            v = fmaf(w11, plane[o11], v);
            sDesc[pt * DESC_PITCH + c] = v;
        }
    }
    __syncthreads();

    // ---- Phase 2: fp32 WMMA whitening GEMM: [16 x 256] x [256 x 16] per wave
    const int wave = t >> 5;
    const int lane = t & 31;
    const int e0   = wave << 4;         // 8 waves cover E = 128
    const int half = lane >> 4;         // 0 -> K0/K1 pair, 1 -> K2/K3 pair (ISA 16x4 layout)
    const int l16  = lane & 15;
    const float bias = wb[e0 + l16];

    v8f acc = {};
    const float* __restrict__ aPtr = &sDesc[l16 * DESC_PITCH + 2 * half];
    const float* __restrict__ bPtr = &ww[(size_t)(e0 + l16) * Cc + 2 * half];
    #pragma unroll 8
    for (int k = 0; k < Cc / 4; ++k) {
        v2f aF = *(const v2f*)(aPtr + 4 * k);   // ds_load_b64, bank-conflict free
        v2f bF = *(const v2f*)(bPtr + 4 * k);   // global_load_b64, L0/L2 hot
        acc = __builtin_amdgcn_wmma_f32_16x16x4_f32(
            /*neg_a=*/false, aF, /*neg_b=*/false, bF,
            /*c_mod=*/(short)0, acc, /*reuse_a=*/false, /*reuse_b=*/false);
    }

    // C/D layout: VGPR r -> (M=r, N=lane) lanes 0-15 ; (M=r+8, N=lane-16) lanes 16-31
    #pragma unroll
    for (int r = 0; r < 8; ++r) {
        const int m = half * 8 + r;
        sOut[m * Ee + e0 + l16] = acc[r] + bias;
    }
    __syncthreads();

    // ---- Phase 3: row-wise L2 norms (2-level reduce)
    {
        const int row = t >> 4;
        const int chunk = t & 15;
        const float* p = &sOut[row * Ee + chunk * 8];
        float s = 0.0f;
        #pragma unroll
        for (int j = 0; j < 8; ++j) s = fmaf(p[j], p[j], s);
        sPart[row * 16 + chunk] = s;
    }
    __syncthreads();
    if (t < 16) {
        float s = 0.0f;
        #pragma unroll
        for (int j = 0; j < 16; ++j) s += sPart[t * 16 + j];
        sRn[t] = 1.0f / fmaxf(sqrtf(s), 1e-12f);
    }
    __syncthreads();

    // ---- Phase 4: scale + coalesced b128 stores
    {
        const int row  = t >> 4;
        const int col0 = (t & 15) * 8;
        const float rn = sRn[row];
        float* __restrict__ dst = out + ((size_t)(b * Nn + n0 + row)) * Ee + col0;
        const float* src = &sOut[row * Ee + col0];
        v4f o0 = { src[0] * rn, src[1] * rn, src[2] * rn, src[3] * rn };
        v4f o1 = { src[4] * rn, src[5] * rn, src[6] * rn, src[7] * rn };
        *(v4f*)(dst)     = o0;
        *(v4f*)(dst + 4) = o1;
    }
}

extern "C" void kernel_launch(void* const* d_in, const int* in_sizes, int n_in,
                              void* d_out, int out_size, void* d_ws, size_t ws_size,
                              hipStream_t stream) {
    const float* x    = (const float*)d_in[0];  // [8,256,128,128]
    const float* kpts = (const float*)d_in[1];  // [8,4096,2]
    const float* ww   = (const float*)d_in[2];  // [128,256]
    const float* wb   = (const float*)d_in[3];  // [128]
    float* out = (float*)d_out;                 // [8,4096,128]

    const int tiles = Bb * (Nn / 16);           // 2048 workgroups, 8 waves each
    localhead_fused_kernel<<<tiles, 256, 0, stream>>>(x, kpts, ww, wb, out);
}